// MultiHeadAttentionLayer_21569325760859
// MI455X (gfx1250) — compile-verified
//
#include <hip/hip_runtime.h>
#include <math.h>

#define N_NODES   50000
#define N_EDGES   800000
#define IN_DIM    128
#define NUM_HEADS 8
#define OUT_DIM   16
#define HD        128   // NUM_HEADS * OUT_DIM

typedef __attribute__((ext_vector_type(2))) float v2f;
typedef __attribute__((ext_vector_type(8))) float v8f;

__device__ __forceinline__ v8f wmma_f32_16x16x4(v2f a, v2f b, v8f c) {
  // 8 args: (neg_a, A, neg_b, B, c_mod, C, reuse_a, reuse_b)
  return __builtin_amdgcn_wmma_f32_16x16x4_f32(false, a, false, b, (short)0, c,
                                               false, false);
}

__global__ void zero_kernel(float* __restrict__ p, int n) {
  int i = blockIdx.x * blockDim.x + threadIdx.x;
  int stride = gridDim.x * blockDim.x;
  for (; i < n; i += stride) p[i] = 0.0f;
}

// One wave computes one 16x16 output tile of out = h @ W + b.
// block = 128 threads (4 waves -> 64 cols), grid = (N/16, 2, 3 matrices)
__global__ __launch_bounds__(128) void node_proj_kernel(
    const float* __restrict__ h,
    const float* __restrict__ Wq, const float* __restrict__ bq,
    const float* __restrict__ Wk, const float* __restrict__ bk,
    const float* __restrict__ Wv, const float* __restrict__ bv,
    float* __restrict__ Qh, float* __restrict__ Kh, float* __restrict__ Vh) {
  const float* W;
  const float* bias;
  float* out;
  if (blockIdx.z == 0) { W = Wq; bias = bq; out = Qh; }
  else if (blockIdx.z == 1) { W = Wk; bias = bk; out = Kh; }
  else { W = Wv; bias = bv; out = Vh; }

  const int lane = threadIdx.x & 31;
  const int wave = threadIdx.x >> 5;
  const int m    = lane & 15;   // A row / B,C col within tile
  const int hi   = lane >> 4;   // half-wave select
  const int rbase = blockIdx.x * 16;
  const int cg    = blockIdx.y * 64 + wave * 16 + m;  // global output column

  v8f c;
#pragma unroll
  for (int i = 0; i < 8; ++i) c[i] = bias[cg];

  const float* __restrict__ Arow = h + (size_t)(rbase + m) * IN_DIM;
#pragma unroll
  for (int k = 0; k < IN_DIM; k += 4) {
    const int ka = k + 2 * hi;
    v2f a, b;
    a.x = Arow[ka];
    a.y = Arow[ka + 1];
    b.x = W[(size_t)ka * HD + cg];
    b.y = W[(size_t)(ka + 1) * HD + cg];
    c = wmma_f32_16x16x4(a, b, c);
  }

#pragma unroll
  for (int i = 0; i < 8; ++i) {
    const int rg = rbase + i + 8 * hi;
    out[(size_t)rg * HD + cg] = c[i];
  }
}

// Fused edge kernel: async e-tile -> LDS, pe GEMM (WMMA), score, e_out (NT),
// exp, scatter. block = 256 threads = 8 waves (wave == head), 16 edges/block.
__global__ __launch_bounds__(256) void edge_kernel(
    const float* __restrict__ e,
    const int* __restrict__ src, const int* __restrict__ dst,
    const float* __restrict__ We, const float* __restrict__ be,
    const float* __restrict__ Qh, const float* __restrict__ Kh,
    const float* __restrict__ Vh,
    float* __restrict__ e_out, float* __restrict__ hout_acc,
    float* __restrict__ z) {
  __shared__ float s_e[16][IN_DIM];          // 8 KB edge-feature tile
  __shared__ float s_sum[NUM_HEADS][16];     // per-head per-edge score sums
  __shared__ float s_sexp[NUM_HEADS][16];    // exp(clamp(sum))
  __shared__ int s_src[16];
  __shared__ int s_dst[16];

  const int tid  = threadIdx.x;
  const int lane = tid & 31;
  const int head = tid >> 5;
  const int m    = lane & 15;
  const int hi   = lane >> 4;
  const int eb   = blockIdx.x * 16;
  const int cg   = head * OUT_DIM + m;       // global feature column

  // --- async DMA of the contiguous 8 KB e-tile into LDS ---------------------
  // 256 threads x 2 x B128 = 8192 B. INST_OFFSET is added to both the LDS
  // and global addresses (ISA 08 §4.4), so one VADDR covers both transfers.
  {
    const unsigned goff = (unsigned)tid * 16u;
    const unsigned ldsa = (unsigned)(uintptr_t)(&s_e[0][0]) + goff;
    const unsigned long long gbase =
        (unsigned long long)(uintptr_t)(e + (size_t)eb * IN_DIM);
    asm volatile(
        "global_load_async_to_lds_b128 %0, %1, %2 offset:0\n\t"
        "global_load_async_to_lds_b128 %0, %1, %2 offset:4096"
        :
        : "v"(ldsa), "v"(goff), "s"(gbase)
        : "memory");
  }

  if (tid < 16) {
    s_src[tid] = src[eb + tid];
    s_dst[tid] = dst[eb + tid];
  }
  if (tid < 128) s_sum[tid >> 4][tid & 15] = 0.0f;

  // bias preload overlaps with the async DMA
  v8f c;
#pragma unroll
  for (int i = 0; i < 8; ++i) c[i] = be[cg];

  asm volatile("s_wait_asynccnt 0x0" ::: "memory");
  __syncthreads();

  // pe tile for this head: [16 edges x 16 cols] = s_e @ We + be
#pragma unroll
  for (int k = 0; k < IN_DIM; k += 4) {
    const int ka = k + 2 * hi;
    v2f a, b;
    a.x = s_e[m][ka];
    a.y = s_e[m][ka + 1];
    b.x = We[(size_t)ka * HD + cg];
    b.y = We[(size_t)(ka + 1) * HD + cg];
    c = wmma_f32_16x16x4(a, b, c);
  }

  // score = Kh[src] * Qh[dst] / sqrt(16) * pe ; write e_out (NT), row sums
#pragma unroll
  for (int i = 0; i < 8; ++i) {
    const int r  = i + 8 * hi;
    const int sN = s_src[r];
    const int dN = s_dst[r];
    const float score =
        Kh[(size_t)sN * HD + cg] * Qh[(size_t)dN * HD + cg] * 0.25f * c[i];
    __builtin_nontemporal_store(score, &e_out[(size_t)(eb + r) * HD + cg]);
    atomicAdd(&s_sum[head][r], score);  // ds_add_f32
  }
  __syncthreads();

  // sexp per (head, edge); accumulate softmax denominator z
  if (tid < 128) {
    const int hh = tid >> 4;
    const int r  = tid & 15;
    float s = s_sum[hh][r];
    s = fminf(fmaxf(s, -5.0f), 5.0f);
    const float ex = __expf(s);
    s_sexp[hh][r] = ex;
    atomicAdd(&z[(size_t)s_dst[r] * NUM_HEADS + hh], ex);
  }
  __syncthreads();

  // numerator scatter: hout_acc[dst] += Vh[src] * sexp
#pragma unroll
  for (int i = 0; i < 8; ++i) {
    const int r  = i + 8 * hi;
    const float ex = s_sexp[head][r];
    const float v  = Vh[(size_t)s_src[r] * HD + cg];
    atomicAdd(&hout_acc[(size_t)s_dst[r] * HD + cg], v * ex);
  }
}

__global__ void finalize_kernel(float* __restrict__ hout,
                                const float* __restrict__ z) {
  const int idx = blockIdx.x * blockDim.x + threadIdx.x;
  if (idx >= N_NODES * HD) return;
  const int node = idx / HD;
  const int head = (idx % HD) / OUT_DIM;
  hout[idx] = hout[idx] / (z[node * NUM_HEADS + head] + 1e-6f);
}

extern "C" void kernel_launch(void* const* d_in, const int* in_sizes, int n_in,
                              void* d_out, int out_size, void* d_ws,
                              size_t ws_size, hipStream_t stream) {
  const float* h  = (const float*)d_in[0];
  const float* e  = (const float*)d_in[1];
  const int* src  = (const int*)d_in[2];
  const int* dst  = (const int*)d_in[3];
  const float* Wq = (const float*)d_in[4];
  const float* bq = (const float*)d_in[5];
  const float* Wk = (const float*)d_in[6];
  const float* bk = (const float*)d_in[7];
  const float* Wv = (const float*)d_in[8];
  const float* bv = (const float*)d_in[9];
  const float* We = (const float*)d_in[10];
  const float* be = (const float*)d_in[11];

  float* hout = (float*)d_out;                       // [N, 128]
  float* eout = hout + (size_t)N_NODES * HD;         // [E, 128]

  float* Qh = (float*)d_ws;                          // [N, 128]
  float* Kh = Qh + (size_t)N_NODES * HD;             // [N, 128]
  float* Vh = Kh + (size_t)N_NODES * HD;             // [N, 128]
  float* z  = Vh + (size_t)N_NODES * HD;             // [N, 8]

  zero_kernel<<<512, 256, 0, stream>>>(hout, N_NODES * HD);
  zero_kernel<<<128, 256, 0, stream>>>(z, N_NODES * NUM_HEADS);

  dim3 gp(N_NODES / 16, 2, 3);
  node_proj_kernel<<<gp, 128, 0, stream>>>(h, Wq, bq, Wk, bk, Wv, bv,
                                           Qh, Kh, Vh);

  edge_kernel<<<N_EDGES / 16, 256, 0, stream>>>(e, src, dst, We, be,
                                                Qh, Kh, Vh, eout, hout, z);

  finalize_kernel<<<(N_NODES * HD + 255) / 256, 256, 0, stream>>>(hout, z);
}